// adder2d_t_49606872269312
// MI455X (gfx1250) — compile-verified
//
#include <hip/hip_runtime.h>
#include <stdint.h>

// ---------------------------------------------------------------------------
// adder2d on MI455X (gfx1250): out[n,f,oh,ow] = Wcol·Xcol - 0.5||W_f||^2 - 0.5||X_l||^2
// GEMM M=128, K=1152, L=28800 via V_WMMA_F32_16X16X4_F32 (fp32 precision).
// Double-buffered LDS; A tile staged with global_load_async_to_lds_b128
// (ASYNCcnt), B tile (padded im2col gather) prefetched through registers.
// LDS tiles are k-pair interleaved (v2f) so every WMMA fragment is a single
// aligned ds_load_b64 into the operand register pair.
// ---------------------------------------------------------------------------

typedef float v2f __attribute__((ext_vector_type(2)));
typedef float v8f __attribute__((ext_vector_type(8)));

#define N_IMG   32
#define C_IN    128
#define H_IN    28
#define F_OUT   128
#define H_OUT   30
#define L_SPAT  (H_OUT * H_OUT)      // 900 spatial positions per image
#define K_TOT   (C_IN * 9)           // 1152
#define PAD     2

#define BL      64                   // output columns per block
#define KT      32                   // K per stage
#define NSTAGE  (K_TOT / KT)         // 36
#define A_STRP  18                   // v2f per filter row (36 floats; conflict-free)
#define B_STRP  80                   // v2f per k-pair row (160 floats; halves 32 banks apart)

// --- kernel 1: ||W_f||^2 ---------------------------------------------------
__global__ __launch_bounds__(128) void wnorm_kernel(const float* __restrict__ W,
                                                    float* __restrict__ wnorm) {
    __shared__ float red[128];
    const int f = blockIdx.x;
    const int t = threadIdx.x;
    float s = 0.f;
    for (int k = t; k < K_TOT; k += 128) {
        float w = W[f * K_TOT + k];
        s += w * w;
    }
    red[t] = s;
    __syncthreads();
    for (int off = 64; off > 0; off >>= 1) {
        if (t < off) red[t] += red[t + off];
        __syncthreads();
    }
    if (t == 0) wnorm[f] = red[0];
}

// --- kernel 2: S[n,h,w] = sum_c x[n,c,h,w]^2 -------------------------------
__global__ __launch_bounds__(256) void chansq_kernel(const float* __restrict__ x,
                                                     float* __restrict__ Ssum) {
    const int idx = blockIdx.x * 256 + threadIdx.x;   // over 32*784
    if (idx >= N_IMG * H_IN * H_IN) return;
    const int n  = idx / (H_IN * H_IN);
    const int hw = idx % (H_IN * H_IN);
    const float* xp = x + (size_t)n * C_IN * H_IN * H_IN + hw;
    float s = 0.f;
    #pragma unroll 4
    for (int c = 0; c < C_IN; ++c) {
        float v = xp[c * H_IN * H_IN];
        s += v * v;
    }
    Ssum[idx] = s;
}

// --- async stage of 128x32 W tile into LDS (k-pair interleaved) ------------
__device__ __forceinline__ void stage_A_async(const float* __restrict__ Wg,
                                              v2f* Abuf, int k0, int t) {
    const int col4  = (t & 7) * 4;     // k offset within tile, multiple of 4
    const int fbase = t >> 3;
    #pragma unroll
    for (int q = 0; q < 4; ++q) {
        const int f = fbase + 32 * q;
        const float* g = Wg + (size_t)f * K_TOT + k0 + col4;
        const unsigned lds = (unsigned)(uintptr_t)(&Abuf[f * A_STRP + (col4 >> 1)]);
        asm volatile("global_load_async_to_lds_b128 %0, %1, off"
                     :: "v"(lds), "v"(g) : "memory");
    }
}

// --- padded im2col gather of 32x64 X tile into registers -------------------
__device__ __forceinline__ void gather_B(const float* __restrict__ ximg,
                                         bool pvalid, int oh, int ow,
                                         int qq, int k0, float br[8]) {
    #pragma unroll
    for (int it = 0; it < 4; ++it) {
        const int kpair = it * 4 + qq;
        #pragma unroll
        for (int h = 0; h < 2; ++h) {
            const int kg = k0 + kpair * 2 + h;
            const int c  = kg / 9;
            const int rs = kg % 9;
            const int r  = rs / 3, s = rs % 3;
            float val = 0.f;
            if (pvalid) {
                const int ih = oh + r - PAD, iw = ow + s - PAD;
                if (ih >= 0 && ih < H_IN && iw >= 0 && iw < H_IN)
                    val = ximg[c * (H_IN * H_IN) + ih * H_IN + iw];
            }
            br[it * 2 + h] = val;
        }
    }
}

__device__ __forceinline__ void store_B(v2f* Bbuf, const float br[8], int qq, int j) {
    #pragma unroll
    for (int it = 0; it < 4; ++it) {
        const int kpair = it * 4 + qq;
        v2f v; v.x = br[it * 2]; v.y = br[it * 2 + 1];
        Bbuf[kpair * B_STRP + j] = v;
    }
}

// --- kernel 3: WMMA implicit-GEMM + quadratic epilogue ---------------------
__global__ __launch_bounds__(256) void adder2d_wmma_kernel(
    const float* __restrict__ x, const float* __restrict__ W,
    const float* __restrict__ wnorm, const float* __restrict__ Ssum,
    float* __restrict__ out)
{
    __shared__ v2f A_s[2][F_OUT * A_STRP];       // 2 x 18432 B
    __shared__ v2f B_s[2][(KT / 2) * B_STRP];    // 2 x 10240 B
    __shared__ float wnorm_s[F_OUT];
    __shared__ float xnorm_s[BL];

    const int t  = threadIdx.x;
    const int n  = blockIdx.y;                   // batch image
    const int p0 = blockIdx.x * BL;              // first spatial position

    const int lane  = t & 31;
    const int wave  = t >> 5;                    // 8 waves
    const int m0    = wave * 16;                 // wave's filter rows
    const int mrow  = m0 + (lane & 15);          // A fragment row
    const int half  = lane >> 4;                 // 0/1: K sub-pair select
    const int nl    = lane & 15;                 // column within 16-wide subtile
    const int khalf = half * 8;                  // C/D row offset per lane half

    // per-thread column (fixed for whole K loop)
    const int j  = t & 63;
    const int qq = t >> 6;                       // k-pair group 0..3
    const int p  = p0 + j;
    const bool pvalid = (p < L_SPAT);
    const int oh = p / H_OUT, ow = p % H_OUT;
    const float* ximg = x + (size_t)n * C_IN * H_IN * H_IN;

    v8f acc[4];
    #pragma unroll
    for (int s4 = 0; s4 < 4; ++s4)
        #pragma unroll
        for (int i = 0; i < 8; ++i) acc[s4][i] = 0.f;

    if (t < F_OUT) wnorm_s[t] = wnorm[t];

    // per-column ||X_l||^2 via 3x3 box sum of channel-square map S
    if (t < BL) {
        float xn = 0.f;
        if (pvalid) {
            const float* Sp = Ssum + (size_t)n * H_IN * H_IN;
            #pragma unroll
            for (int r = 0; r < 3; ++r) {
                const int ih = oh + r - PAD;
                if (ih < 0 || ih >= H_IN) continue;
                #pragma unroll
                for (int s = 0; s < 3; ++s) {
                    const int iw = ow + s - PAD;
                    if (iw >= 0 && iw < H_IN) xn += Sp[ih * H_IN + iw];
                }
            }
        }
        xnorm_s[t] = xn;
    }

    // ---- prologue: stage tile 0 ----
    float br[8];
    stage_A_async(W, A_s[0], 0, t);
    gather_B(ximg, pvalid, oh, ow, qq, 0, br);
    store_B(B_s[0], br, qq, j);
    asm volatile("s_wait_asynccnt 0x0" ::: "memory");
    __syncthreads();

    // ---- K loop: 36 double-buffered stages, one barrier each ----
    for (int st = 0; st < NSTAGE; ++st) {
        const int cur = st & 1;
        const bool more = (st + 1 < NSTAGE);

        if (more) {
            stage_A_async(W, A_s[cur ^ 1], (st + 1) * KT, t);   // async -> spare buf
            gather_B(ximg, pvalid, oh, ow, qq, (st + 1) * KT, br);
        }

        const v2f* __restrict__ Ab = A_s[cur];
        const v2f* __restrict__ Bb = B_s[cur];
        #pragma unroll
        for (int kk = 0; kk < 8; ++kk) {
            const v2f a = Ab[mrow * A_STRP + kk * 2 + half];
            #pragma unroll
            for (int s4 = 0; s4 < 4; ++s4) {
                const v2f b = Bb[(kk * 2 + half) * B_STRP + s4 * 16 + nl];
                acc[s4] = __builtin_amdgcn_wmma_f32_16x16x4_f32(
                    /*neg_a=*/false, a, /*neg_b=*/false, b,
                    /*c_mod=*/(short)0, acc[s4],
                    /*reuse_a=*/false, /*reuse_b=*/false);
            }
        }

        if (more) {
            store_B(B_s[cur ^ 1], br, qq, j);
            asm volatile("s_wait_asynccnt 0x0" ::: "memory");
        }
        __syncthreads();
    }

    // ---- epilogue: subtract 0.5||W_f||^2 + 0.5||X_l||^2, scatter NCHW ----
    #pragma unroll
    for (int s4 = 0; s4 < 4; ++s4) {
        const int jj = s4 * 16 + nl;
        const int pp = p0 + jj;
        if (pp < L_SPAT) {
            const int ohh = pp / H_OUT, oww = pp % H_OUT;
            const float xn = 0.5f * xnorm_s[jj];
            #pragma unroll
            for (int i = 0; i < 8; ++i) {
                const int f = m0 + khalf + i;
                const float val = acc[s4][i] - 0.5f * wnorm_s[f] - xn;
                out[(((size_t)n * F_OUT + f) * H_OUT + ohh) * H_OUT + oww] = val;
            }
        }
    }
}

extern "C" void kernel_launch(void* const* d_in, const int* in_sizes, int n_in,
                              void* d_out, int out_size, void* d_ws, size_t ws_size,
                              hipStream_t stream) {
    const float* x = (const float*)d_in[0];   // (32,128,28,28) f32
    const float* W = (const float*)d_in[1];   // (128,128,3,3)  f32
    float* out = (float*)d_out;               // (32,128,30,30) f32

    float* wnorm = (float*)d_ws;              // 128 floats
    float* Ssum  = wnorm + F_OUT;             // 32*28*28 floats
    (void)in_sizes; (void)n_in; (void)out_size; (void)ws_size;

    wnorm_kernel<<<F_OUT, 128, 0, stream>>>(W, wnorm);

    const int nS = N_IMG * H_IN * H_IN;
    chansq_kernel<<<(nS + 255) / 256, 256, 0, stream>>>(x, Ssum);

    dim3 grid((L_SPAT + BL - 1) / BL, N_IMG); // (15, 32)
    adder2d_wmma_kernel<<<grid, 256, 0, stream>>>(x, W, wnorm, Ssum, out);
}